// AttentionBlock_3375844294910
// MI455X (gfx1250) — compile-verified
//
#include <hip/hip_runtime.h>
#include <cstddef>
#include <cstdint>

typedef __attribute__((ext_vector_type(2))) float v2f;
typedef __attribute__((ext_vector_type(8))) float v8f;

#define F_DIM 128
#define C_DIM 64
#define ROWS_PER_WG 160   // 10 waves x 16 rows
#define NTHREADS 320
#define BN_EPS 1e-5f

// ---- workspace layout (float offsets) ----
// [0..255]    raw stats: sum_g[64], ss_g[64], sum_x[64], ss_x[64]
// [256..447]  coefs: sg[64], sx[64], off[64]
// [512..513]  y stats: sum_y, ss_y
// [516..517]  a, b  (psi = sigmoid(y*a + b))
// [1024..]    y[N]
#define WS_STATS 0
#define WS_COEF  256
#define WS_YSTAT 512
#define WS_AB    516
#define WS_Y     1024

// ---------------------------------------------------------------------------
// Shared GEMM micro-kernel: one wave computes a 16x64 tile of  zg = gate@Wg
// and zx = skip@Wx using V_WMMA_F32_16X16X4_F32, accumulating over K=128.
//
// f32 16x16x4 A layout (ISA 7.12.2): lanes 0-15 hold rows M=0..15 with
// K=0 (v0), K=1 (v1); lanes 16-31 hold K=2 (v0), K=3 (v1).
// B (4x16) mirrored: v0 = row K(+0 / +2 per half), v1 = row K+1(+2).
// D: v8f, lane L holds col N=L%16, rows r + 8*(L/16) for r=VGPR index.
// ---------------------------------------------------------------------------
__device__ __forceinline__ void gemm_tile(
    const float* __restrict__ gate, const float* __restrict__ skip,
    const float* sWg, const float* sWx,
    int rowbase, int lane, v8f (&accG)[4], v8f (&accX)[4])
{
    const int lo = lane & 15;
    const int hi = lane >> 4;
    const float* grow = gate + (size_t)(rowbase + lo) * F_DIM + 2 * hi;
    const float* srow = skip + (size_t)(rowbase + lo) * F_DIM + 2 * hi;
    const int wofs = 2 * hi * C_DIM;   // K row offset for this half-wave

    for (int k = 0; k < F_DIM; k += 4) {
        v2f ag = *(const v2f*)(grow + k);
        v2f as = *(const v2f*)(srow + k);
        const float* wg = sWg + k * C_DIM + wofs;
        const float* wx = sWx + k * C_DIM + wofs;
#pragma unroll
        for (int ct = 0; ct < 4; ++ct) {
            const int cc = ct * 16 + lo;
            v2f bg, bx;
            bg.x = wg[cc];         bg.y = wg[C_DIM + cc];
            bx.x = wx[cc];         bx.y = wx[C_DIM + cc];
            accG[ct] = __builtin_amdgcn_wmma_f32_16x16x4_f32(
                false, ag, false, bg, (short)0, accG[ct], false, false);
            accX[ct] = __builtin_amdgcn_wmma_f32_16x16x4_f32(
                false, as, false, bx, (short)0, accX[ct], false, false);
        }
    }
}

__device__ __forceinline__ void load_weights(const float* __restrict__ Wg,
                                             const float* __restrict__ Wx,
                                             float* sWg, float* sWx, int tid)
{
    const float4* g4 = (const float4*)Wg;
    const float4* x4 = (const float4*)Wx;
    float4* sg4 = (float4*)sWg;
    float4* sx4 = (float4*)sWx;
    for (int i = tid; i < (F_DIM * C_DIM) / 4; i += NTHREADS) {
        sg4[i] = g4[i];
        sx4[i] = x4[i];
    }
}

// ---------------------------------------------------------------------------
// Kernel 0: zero the small stats region of the workspace (ws is poisoned).
// ---------------------------------------------------------------------------
__global__ void k_init(float* ws)
{
    ws[threadIdx.x] = 0.0f;   // 1024 threads cover [0..1023]
}

// ---------------------------------------------------------------------------
// Kernel 1: GEMMs + per-channel sum / sum-of-squares for the two BNs.
// ---------------------------------------------------------------------------
__global__ __launch_bounds__(NTHREADS)
void k_gemm_stats(const float* __restrict__ gate, const float* __restrict__ skip,
                  const float* __restrict__ Wg, const float* __restrict__ Wx,
                  float* __restrict__ stats, int Nt)
{
    __shared__ float sWg[F_DIM * C_DIM];
    __shared__ float sWx[F_DIM * C_DIM];
    __shared__ float sStat[256];

    const int tid = threadIdx.x;
    load_weights(Wg, Wx, sWg, sWx, tid);
    if (tid < 256) sStat[tid] = 0.0f;
    __syncthreads();

    const int wave = tid >> 5;
    const int lane = tid & 31;
    const int rowbase = blockIdx.x * ROWS_PER_WG + wave * 16;
    const bool active = rowbase < Nt;   // wave-uniform

    if (active) {
        v8f accG[4], accX[4];
#pragma unroll
        for (int ct = 0; ct < 4; ++ct) {
            accG[ct] = (v8f){0,0,0,0,0,0,0,0};
            accX[ct] = (v8f){0,0,0,0,0,0,0,0};
        }
        gemm_tile(gate, skip, sWg, sWx, rowbase, lane, accG, accX);

        const int lo = lane & 15;
#pragma unroll
        for (int ct = 0; ct < 4; ++ct) {
            float sg = 0.f, ssg = 0.f, sx = 0.f, ssx = 0.f;
#pragma unroll
            for (int r = 0; r < 8; ++r) {
                float g = accG[ct][r], x = accX[ct][r];
                sg += g; ssg += g * g; sx += x; ssx += x * x;
            }
            // fold rows 8..15 (other half-wave, same channel)
            sg  += __shfl_xor(sg, 16);
            ssg += __shfl_xor(ssg, 16);
            sx  += __shfl_xor(sx, 16);
            ssx += __shfl_xor(ssx, 16);
            if (lane < 16) {
                const int c = ct * 16 + lo;
                atomicAdd(&sStat[c], sg);
                atomicAdd(&sStat[64 + c], ssg);
                atomicAdd(&sStat[128 + c], sx);
                atomicAdd(&sStat[192 + c], ssx);
            }
        }
    }
    __syncthreads();
    if (tid < 256) atomicAdd(&stats[tid], sStat[tid]);
}

// ---------------------------------------------------------------------------
// Kernel 2: finalize BN coefficients for both branches.
// g1 + x1 = zg*sg + zx*sx + off   (biases bg/bx cancel inside BN)
// ---------------------------------------------------------------------------
__global__ void k_finalize1(const float* __restrict__ stats,
                            const float* __restrict__ gamma_g, const float* __restrict__ beta_g,
                            const float* __restrict__ gamma_x, const float* __restrict__ beta_x,
                            float* __restrict__ coef, int Nt)
{
    const int c = threadIdx.x;   // 64 threads
    const float invN = 1.0f / (float)Nt;
    const float mg = stats[c] * invN;
    const float vg = stats[64 + c] * invN - mg * mg;
    const float mx = stats[128 + c] * invN;
    const float vx = stats[192 + c] * invN - mx * mx;
    const float sg = gamma_g[c] * rsqrtf(vg + BN_EPS);
    const float sx = gamma_x[c] * rsqrtf(vx + BN_EPS);
    coef[c]        = sg;
    coef[64 + c]   = sx;
    coef[128 + c]  = (beta_g[c] - mg * sg) + (beta_x[c] - mx * sx);
}

// ---------------------------------------------------------------------------
// Kernel 3: recompute GEMMs, apply BN + relu, dot with Wpsi -> y[n];
// accumulate y's sum / sum-of-squares for the final BN.
// ---------------------------------------------------------------------------
__global__ __launch_bounds__(NTHREADS)
void k_psi(const float* __restrict__ gate, const float* __restrict__ skip,
           const float* __restrict__ Wg, const float* __restrict__ Wx,
           const float* __restrict__ coef, const float* __restrict__ Wpsi,
           float* __restrict__ y, float* __restrict__ ystats, int Nt)
{
    __shared__ float sWg[F_DIM * C_DIM];
    __shared__ float sWx[F_DIM * C_DIM];
    __shared__ float sY[2];

    const int tid = threadIdx.x;
    load_weights(Wg, Wx, sWg, sWx, tid);
    if (tid < 2) sY[tid] = 0.0f;
    __syncthreads();

    const int wave = tid >> 5;
    const int lane = tid & 31;
    const int rowbase = blockIdx.x * ROWS_PER_WG + wave * 16;
    const bool active = rowbase < Nt;

    if (active) {
        v8f accG[4], accX[4];
#pragma unroll
        for (int ct = 0; ct < 4; ++ct) {
            accG[ct] = (v8f){0,0,0,0,0,0,0,0};
            accX[ct] = (v8f){0,0,0,0,0,0,0,0};
        }
        gemm_tile(gate, skip, sWg, sWx, rowbase, lane, accG, accX);

        const int lo = lane & 15;
        float sgv[4], sxv[4], ofv[4], wpv[4];
#pragma unroll
        for (int ct = 0; ct < 4; ++ct) {
            const int c = ct * 16 + lo;
            sgv[ct] = coef[c];
            sxv[ct] = coef[64 + c];
            ofv[ct] = coef[128 + c];
            wpv[ct] = Wpsi[c];
        }

        float t[8];
#pragma unroll
        for (int r = 0; r < 8; ++r) t[r] = 0.0f;
#pragma unroll
        for (int ct = 0; ct < 4; ++ct) {
#pragma unroll
            for (int r = 0; r < 8; ++r) {
                float p = accG[ct][r] * sgv[ct] + accX[ct][r] * sxv[ct] + ofv[ct];
                p = fmaxf(p, 0.0f);            // relu
                t[r] += p * wpv[ct];           // partial dot with Wpsi
            }
        }
        // reduce across the 16 lanes of each half-wave: after this every lane
        // in a half holds y[rowbase + r (+8 for the upper half)]
#pragma unroll
        for (int r = 0; r < 8; ++r) {
            t[r] += __shfl_xor(t[r], 1);
            t[r] += __shfl_xor(t[r], 2);
            t[r] += __shfl_xor(t[r], 4);
            t[r] += __shfl_xor(t[r], 8);
        }
        if (lane == 0 || lane == 16) {
            const int rb = rowbase + (lane >> 4) * 8;
            float s = 0.f, ss = 0.f;
#pragma unroll
            for (int r = 0; r < 8; ++r) {
                y[rb + r] = t[r];
                s += t[r];
                ss += t[r] * t[r];
            }
            atomicAdd(&sY[0], s);
            atomicAdd(&sY[1], ss);
        }
    }
    __syncthreads();
    if (tid == 0) {
        atomicAdd(&ystats[0], sY[0]);
        atomicAdd(&ystats[1], sY[1]);
    }
}

// ---------------------------------------------------------------------------
// Kernel 4: finalize the scalar BN:  psi = sigmoid(y*a + b)  (bpsi cancels)
// ---------------------------------------------------------------------------
__global__ void k_finalize2(const float* __restrict__ ystats,
                            const float* __restrict__ gamma_psi,
                            const float* __restrict__ beta_psi,
                            float* __restrict__ ab, int Nt)
{
    const float invN = 1.0f / (float)Nt;
    const float m = ystats[0] * invN;
    const float v = ystats[1] * invN - m * m;
    const float a = gamma_psi[0] * rsqrtf(v + BN_EPS);
    ab[0] = a;
    ab[1] = beta_psi[0] - m * a;
}

// ---------------------------------------------------------------------------
// Kernel 5: out[n,f] = skip[n,f] * sigmoid(y[n]*a + b)   (float4 streaming)
// ---------------------------------------------------------------------------
__global__ void k_out(const float* __restrict__ skip, const float* __restrict__ y,
                      const float* __restrict__ ab, float* __restrict__ out, int nvec)
{
    const int i = blockIdx.x * blockDim.x + threadIdx.x;
    if (i >= nvec) return;
    const int n = i >> 5;                     // 32 float4 per row (F=128)
    const float t = y[n] * ab[0] + ab[1];
    const float p = 1.0f / (1.0f + __expf(-t));
    float4 v = ((const float4*)skip)[i];
    v.x *= p; v.y *= p; v.z *= p; v.w *= p;
    ((float4*)out)[i] = v;
}

// ---------------------------------------------------------------------------
extern "C" void kernel_launch(void* const* d_in, const int* in_sizes, int n_in,
                              void* d_out, int out_size, void* d_ws, size_t ws_size,
                              hipStream_t stream)
{
    const float* gate      = (const float*)d_in[0];
    const float* skip      = (const float*)d_in[1];
    const float* Wg        = (const float*)d_in[2];
    // d_in[3] = bg       (cancels inside BN)
    const float* gamma_g   = (const float*)d_in[4];
    const float* beta_g    = (const float*)d_in[5];
    const float* Wx        = (const float*)d_in[6];
    // d_in[7] = bx       (cancels inside BN)
    const float* gamma_x   = (const float*)d_in[8];
    const float* beta_x    = (const float*)d_in[9];
    const float* Wpsi      = (const float*)d_in[10];
    // d_in[11] = bpsi    (cancels inside BN)
    const float* gamma_psi = (const float*)d_in[12];
    const float* beta_psi  = (const float*)d_in[13];
    float* out = (float*)d_out;

    const int Nt = in_sizes[0] / F_DIM;       // 1,000,000

    float* ws     = (float*)d_ws;
    float* stats  = ws + WS_STATS;
    float* coef   = ws + WS_COEF;
    float* ystats = ws + WS_YSTAT;
    float* ab     = ws + WS_AB;
    float* yv     = ws + WS_Y;

    const int nb = (Nt + ROWS_PER_WG - 1) / ROWS_PER_WG;   // 6250
    const int nvec = Nt * (F_DIM / 4);                     // 32M float4
    const int nb_out = (nvec + 255) / 256;

    k_init<<<1, 1024, 0, stream>>>(ws);
    k_gemm_stats<<<nb, NTHREADS, 0, stream>>>(gate, skip, Wg, Wx, stats, Nt);
    k_finalize1<<<1, 64, 0, stream>>>(stats, gamma_g, beta_g, gamma_x, beta_x, coef, Nt);
    k_psi<<<nb, NTHREADS, 0, stream>>>(gate, skip, Wg, Wx, coef, Wpsi, yv, ystats, Nt);
    k_finalize2<<<1, 1, 0, stream>>>(ystats, gamma_psi, beta_psi, ab, Nt);
    k_out<<<nb_out, 256, 0, stream>>>(skip, yv, ab, out, nvec);
}